// MotionGenerator_18116172054819
// MI455X (gfx1250) — compile-verified
//
#include <hip/hip_runtime.h>

typedef float v2f __attribute__((ext_vector_type(2)));
typedef float v8f __attribute__((ext_vector_type(8)));

constexpr int N_NODES = 100000;   // nodes (== 6250 * 16)
constexpr int E_EDGES = 1600000;  // edges (self-loops handled analytically)
constexpr int G_GRAPHS = 128;
constexpr int IN_CH = 128;
constexpr int HID = 64;
constexpr int OUT_CH = 7;
constexpr int N_STACK = 9;

#define CDIV(a, b) (((a) + (b) - 1) / (b))

// ---------------- degree / normalization precompute ----------------

__global__ void init_deg_k(float* __restrict__ deg) {
  int i = blockIdx.x * 256 + threadIdx.x;
  if (i < N_NODES) deg[i] = 1.0f;  // self-loop contributes 1 to in-degree
}

__global__ void deg_accum_k(const int* __restrict__ dst, float* __restrict__ deg) {
  int e = blockIdx.x * 256 + threadIdx.x;
  if (e < E_EDGES) atomicAdd(&deg[dst[e]], 1.0f);
}

__global__ void dinv_k(float* __restrict__ deg) {
  int i = blockIdx.x * 256 + threadIdx.x;
  if (i < N_NODES) deg[i] = rsqrtf(deg[i]);  // deg >= 1 always (self-loop)
}

__global__ void norm_k(const int* __restrict__ src, const int* __restrict__ dst,
                       const float* __restrict__ dinv, float* __restrict__ nrm) {
  int e = blockIdx.x * 256 + threadIdx.x;
  if (e < E_EDGES) nrm[e] = dinv[src[e]] * dinv[dst[e]];
}

// ---------------- dense GEMM: M = Hin(N x K) @ W(K x 64), f32 WMMA ----------------
// Block = 128 threads = 4 waves. Block handles 16 rows; wave w handles cols [16w,16w+16).
// V_WMMA_F32_16X16X4_F32 fragment layouts (wave32):
//   A(16x4):  lane L -> row L&15, K = 2*(L>>4) + v      (v = frag element 0..1)
//   B(4x16):  lane L -> col L&15, K = 2*(L>>4) + v
//   C/D:      lane L, elem v -> row 8*(L>>4) + v, col L&15
__global__ void gcn_gemm_k(const float* __restrict__ Hin, const float* __restrict__ W,
                           float* __restrict__ M, int K) {
  const int lane = threadIdx.x & 31;
  const int wave = threadIdx.x >> 5;
  const int r0   = blockIdx.x << 4;       // 16-row tile
  const int arow = lane & 15;
  const int koff = (lane >> 4) << 1;      // 0 or 2
  const int bcol = lane & 15;
  const int col0 = wave << 4;

  v8f c = {};
  const float* __restrict__ aptr = Hin + (size_t)(r0 + arow) * K + koff;

#if __has_builtin(__builtin_amdgcn_wmma_f32_16x16x4_f32)
  for (int kk = 0; kk < K; kk += 4) {
    v2f a, b;
    a.x = aptr[kk];
    a.y = aptr[kk + 1];
    b.x = W[(size_t)(kk + koff) * HID + col0 + bcol];
    b.y = W[(size_t)(kk + koff + 1) * HID + col0 + bcol];
    c = __builtin_amdgcn_wmma_f32_16x16x4_f32(false, a, false, b, (short)0, c,
                                              false, false);
  }
#else
  // scalar fallback with identical output mapping (signals wmma=0 in histogram)
  {
    const int crow0 = r0 + ((lane >> 4) << 3);
    for (int kk = 0; kk < K; ++kk) {
      float bv = W[(size_t)kk * HID + col0 + bcol];
#pragma unroll
      for (int v = 0; v < 8; ++v)
        c[v] = fmaf(Hin[(size_t)(crow0 + v) * K + kk], bv, c[v]);
    }
  }
#endif

#pragma unroll
  for (int v = 0; v < 8; ++v) {
    int row = r0 + ((lane >> 4) << 3) + v;
    M[(size_t)row * HID + col0 + bcol] = c[v];
  }
}

// ---------------- aggregation ----------------

// agg[i,:] = dinv[i]^2 * M[i,:]   (self-loop term; also serves as the zero-init)
__global__ void self_loop_k(const float* __restrict__ M, const float* __restrict__ dinv,
                            float* __restrict__ agg) {
  int idx = blockIdx.x * 256 + threadIdx.x;
  if (idx < N_NODES * HID) {
    float d = dinv[idx >> 6];
    agg[idx] = d * d * M[idx];
  }
}

// 16 threads per edge; each handles 4 contiguous floats of the 64-wide row.
__global__ void edge_scatter_k(const int* __restrict__ src, const int* __restrict__ dst,
                               const float* __restrict__ nrm, const float* __restrict__ M,
                               float* __restrict__ agg) {
  int idx = blockIdx.x * 256 + threadIdx.x;  // E*16 = 25.6M < 2^31
  int e = idx >> 4;
  int c = (idx & 15) << 2;
  if (e < E_EDGES) {
    int s = src[e], d = dst[e];
    float w = nrm[e];
    const float4 m = *(const float4*)(M + (size_t)s * HID + c);
    float* a = agg + (size_t)d * HID + c;
    atomicAdd(a + 0, w * m.x);
    atomicAdd(a + 1, w * m.y);
    atomicAdd(a + 2, w * m.z);
    atomicAdd(a + 3, w * m.w);
  }
}

__global__ void bias_relu_k(float* __restrict__ h, const float* __restrict__ b) {
  int idx = blockIdx.x * 256 + threadIdx.x;
  if (idx < N_NODES * HID) {
    float v = h[idx] + b[idx & 63];
    h[idx] = v > 0.0f ? v : 0.0f;
  }
}

// ---------------- pooling + final linear ----------------

__global__ void zero_k(float* __restrict__ p, int n) {
  int i = blockIdx.x * 256 + threadIdx.x;
  if (i < n) p[i] = 0.0f;
}

__global__ void pool_scatter_k(const float* __restrict__ h, const int* __restrict__ batch,
                               float* __restrict__ pool, float* __restrict__ cnt) {
  int idx = blockIdx.x * 256 + threadIdx.x;
  if (idx < N_NODES * HID) {
    int i = idx >> 6, c = idx & 63;
    int g = batch[i];
    atomicAdd(&pool[g * HID + c], h[idx]);
    if (c == 0) atomicAdd(&cnt[g], 1.0f);
  }
}

__global__ void final_lin_k(const float* __restrict__ pool, const float* __restrict__ cnt,
                            const float* __restrict__ Wlin, const float* __restrict__ blin,
                            float* __restrict__ out) {
  int idx = blockIdx.x * 256 + threadIdx.x;
  if (idx < G_GRAPHS * OUT_CH) {
    int g = idx / OUT_CH, o = idx - g * OUT_CH;
    float acc = 0.0f;
#pragma unroll
    for (int j = 0; j < HID; ++j)
      acc = fmaf(pool[g * HID + j], Wlin[j * OUT_CH + o], acc);
    float c = cnt[g];
    c = c < 1.0f ? 1.0f : c;
    out[idx] = acc / c + blin[o];
  }
}

// ---------------- host orchestration ----------------

extern "C" void kernel_launch(void* const* d_in, const int* in_sizes, int n_in,
                              void* d_out, int out_size, void* d_ws, size_t ws_size,
                              hipStream_t stream) {
  const float* x    = (const float*)d_in[0];           // [N, IN]
  const int*   edge = (const int*)d_in[1];             // [2, E]
  const int*   src  = edge;
  const int*   dst  = edge + E_EDGES;
  const int*   batch = (const int*)d_in[2];            // [N]
  const float* W0   = (const float*)d_in[3];           // [IN, H]
  const float* b0   = (const float*)d_in[4];           // [H]
  const float* Ws   = (const float*)d_in[5];           // [9, H, H]
  const float* bs   = (const float*)d_in[6];           // [9, H]
  const float* Wlin = (const float*)d_in[7];           // [H, OUT]
  const float* blin = (const float*)d_in[8];           // [OUT]
  float* out = (float*)d_out;                          // [G, OUT]

  // workspace layout (floats), 64-float aligned sections
  float* base = (float*)d_ws;
  size_t off = 0;
  float* dinv = base + off; off += 100096;               // N rounded up to 128
  float* nrm  = base + off; off += (size_t)E_EDGES;
  float* buf0 = base + off; off += (size_t)N_NODES * HID; // h_in / agg / h_out
  float* bufM = base + off; off += (size_t)N_NODES * HID; // M = h @ W
  float* pool = base + off; off += (size_t)G_GRAPHS * HID;
  float* cnt  = base + off; off += G_GRAPHS;
  (void)off; (void)ws_size; (void)n_in; (void)in_sizes; (void)out_size;

  const int NH = N_NODES * HID;

  // normalization coefficients (once, reused by all 10 layers)
  init_deg_k<<<CDIV(N_NODES, 256), 256, 0, stream>>>(dinv);
  deg_accum_k<<<CDIV(E_EDGES, 256), 256, 0, stream>>>(dst, dinv);
  dinv_k<<<CDIV(N_NODES, 256), 256, 0, stream>>>(dinv);
  norm_k<<<CDIV(E_EDGES, 256), 256, 0, stream>>>(src, dst, dinv, nrm);

  // 10 GCN layers
  for (int l = 0; l < 1 + N_STACK; ++l) {
    const float* hin = (l == 0) ? x : buf0;
    const float* Wl  = (l == 0) ? W0 : (Ws + (size_t)(l - 1) * HID * HID);
    const float* bl  = (l == 0) ? b0 : (bs + (size_t)(l - 1) * HID);
    const int K      = (l == 0) ? IN_CH : HID;

    gcn_gemm_k<<<N_NODES / 16, 128, 0, stream>>>(hin, Wl, bufM, K);
    self_loop_k<<<CDIV(NH, 256), 256, 0, stream>>>(bufM, dinv, buf0);
    edge_scatter_k<<<CDIV(E_EDGES * 16, 256), 256, 0, stream>>>(src, dst, nrm, bufM, buf0);
    bias_relu_k<<<CDIV(NH, 256), 256, 0, stream>>>(buf0, bl);
  }

  // global mean pool + final linear
  zero_k<<<CDIV(G_GRAPHS * HID + G_GRAPHS, 256), 256, 0, stream>>>(pool, G_GRAPHS * HID + G_GRAPHS);
  pool_scatter_k<<<CDIV(NH, 256), 256, 0, stream>>>(buf0, batch, pool, cnt);
  final_lin_k<<<CDIV(G_GRAPHS * OUT_CH, 256), 256, 0, stream>>>(pool, cnt, Wlin, blin, out);
}